// FEAf_32057635897485
// MI455X (gfx1250) — compile-verified
//
#include <hip/hip_runtime.h>
#include <hip/hip_bf16.h>

// FEDformer Fourier cross-attention for MI455X (gfx1250).
// All big GEMMs use V_WMMA_F32_16X16X4_F32 (fp32 tensor path, wave32) with
// register-staged double buffering (global->VGPR prefetch overlaps WMMA).
// B=8, T=4096, H=16, D=64, M=256 modes.

typedef float v2f __attribute__((ext_vector_type(2)));
typedef float v8f __attribute__((ext_vector_type(8)));

#define ZERO8 {0.f,0.f,0.f,0.f,0.f,0.f,0.f,0.f}

static constexpr int   Bn = 8, Tn = 4096, Hn = 16, Dn = 64, Mn = 256;
static constexpr float TWO_PI_OVER_T = 1.5339807878856412e-03f;  // 2*pi/4096
static constexpr float SCALE2 = 1.3877787807814457e-17f;         // (1/2^28)^2

// ---------------- WMMA helpers ----------------------------------------------

__device__ __forceinline__ v8f wmma4(v2f a, v2f b, v8f c) {
  // D = A(16x4,f32) * B(4x16,f32) + C(16x16,f32). NEG[0]/NEG[1] must be 0 for F32.
  return __builtin_amdgcn_wmma_f32_16x16x4_f32(false, a, false, b, (short)0, c,
                                               false, false);
}

// LDS tiles are stored [k][m] (or [k][n]) with padded row stride 68.
// A-frag layout (16x4 f32): lanes<16 hold K=k,k+1; lanes>=16 hold K=k+2,k+3; M=lane%16.
__device__ __forceinline__ v2f frag_ld(const float* s, int kk, int moff) {
  const int lane = threadIdx.x & 31;
  const int half = lane >> 4;
  const int lm   = lane & 15;
  v2f f;
  f.x = s[(kk + 2 * half    ) * 68 + moff + lm];
  f.y = s[(kk + 2 * half + 1) * 68 + moff + lm];
  return f;
}

// ---------------- register-staged tile movement (256 threads) ---------------

struct F4x4 { float4 a, b, c, d; };   // 64x64 tile: 4 float4 per thread
struct F4x2 { float4 a, b; };         // 32x64 tile: 2 float4 per thread

// 64-row fetch, rows rg+{0,16,32,48}, 16B columns (t&15)*4.
__device__ __forceinline__ F4x4 fetch_64(const float* src, int ld) {
  const int t = threadIdx.x;
  const int rg = t >> 4;
  const int cq = (t & 15) << 2;
  F4x4 r;
  r.a = *(const float4*)(src + (size_t)(rg +  0) * ld + cq);
  r.b = *(const float4*)(src + (size_t)(rg + 16) * ld + cq);
  r.c = *(const float4*)(src + (size_t)(rg + 32) * ld + cq);
  r.d = *(const float4*)(src + (size_t)(rg + 48) * ld + cq);
  return r;
}

// stash as dst[k][m] when source rows were k (direct): contiguous b128 LDS stores
__device__ __forceinline__ void stash64_direct(float* dst, const F4x4& r) {
  const int t = threadIdx.x;
  const int rg = t >> 4;
  const int cq = (t & 15) << 2;
  *(float4*)(dst + (rg +  0) * 68 + cq) = r.a;
  *(float4*)(dst + (rg + 16) * 68 + cq) = r.b;
  *(float4*)(dst + (rg + 32) * 68 + cq) = r.c;
  *(float4*)(dst + (rg + 48) * 68 + cq) = r.d;
}

// stash as dst[k][m] when source rows were m (transpose): scalar scatter
__device__ __forceinline__ void stash64_trans(float* dst, const F4x4& r) {
  const int t = threadIdx.x;
  const int rg = t >> 4;
  const int kq = (t & 15) << 2;
  const float4 vv[4] = {r.a, r.b, r.c, r.d};
#pragma unroll
  for (int i = 0; i < 4; ++i) {
    const int m = i * 16 + rg;
    dst[(kq + 0) * 68 + m] = vv[i].x;
    dst[(kq + 1) * 68 + m] = vv[i].y;
    dst[(kq + 2) * 68 + m] = vv[i].z;
    dst[(kq + 3) * 68 + m] = vv[i].w;
  }
}

// 32(k) x 64(m) tile, source rows are k
__device__ __forceinline__ F4x2 fetch_32_direct(const float* src, int ld) {
  const int t = threadIdx.x;
  const int rg = t >> 4;
  const int cq = (t & 15) << 2;
  F4x2 r;
  r.a = *(const float4*)(src + (size_t)(rg +  0) * ld + cq);
  r.b = *(const float4*)(src + (size_t)(rg + 16) * ld + cq);
  return r;
}

__device__ __forceinline__ void stash32_direct(float* dst, const F4x2& r) {
  const int t = threadIdx.x;
  const int rg = t >> 4;
  const int cq = (t & 15) << 2;
  *(float4*)(dst + (rg +  0) * 68 + cq) = r.a;
  *(float4*)(dst + (rg + 16) * 68 + cq) = r.b;
}

// 32(k) x 64(m) tile, source rows are m (64 rows, 32 contiguous k)
__device__ __forceinline__ F4x2 fetch_32_trans(const float* src, int ld) {
  const int t = threadIdx.x;
  const int rg = t >> 3;          // 0..31
  const int kq = (t & 7) << 2;    // 0..28
  F4x2 r;
  r.a = *(const float4*)(src + (size_t)(rg +  0) * ld + kq);
  r.b = *(const float4*)(src + (size_t)(rg + 32) * ld + kq);
  return r;
}

__device__ __forceinline__ void stash32_trans(float* dst, const F4x2& r) {
  const int t = threadIdx.x;
  const int rg = t >> 3;
  const int kq = (t & 7) << 2;
  const float4 vv[2] = {r.a, r.b};
#pragma unroll
  for (int i = 0; i < 2; ++i) {
    const int m = i * 32 + rg;
    dst[(kq + 0) * 68 + m] = vv[i].x;
    dst[(kq + 1) * 68 + m] = vv[i].y;
    dst[(kq + 2) * 68 + m] = vv[i].z;
    dst[(kq + 3) * 68 + m] = vv[i].w;
  }
}

// ---------------- basis construction ----------------------------------------

// Forward DFT basis fb[t][n], n<256: cos(th), n>=256: -sin(th); th = 2*pi*idx[n%256]*t/T
__global__ __launch_bounds__(256) void build_fbasis_kernel(const int* __restrict__ idx,
                                                           float* __restrict__ fb) {
  const int id = blockIdx.x * 256 + threadIdx.x;  // t*512 + n
  const int t = id >> 9;
  const int n = id & 511;
  const int f = idx[n & 255];
  const int p = (f * t) & (Tn - 1);               // exact: f*t < 2^24
  const float th = (float)p * TWO_PI_OVER_T;
  fb[id] = (n < Mn) ? cosf(th) : -sinf(th);
}

// Inverse basis ib[n][t]: row n<256: (c/T)cos(th), n>=256: -(c/T)sin(th); c = (idx==0?1:2)
__global__ __launch_bounds__(256) void build_ibasis_kernel(const int* __restrict__ idx,
                                                           float* __restrict__ ib) {
  const int id = blockIdx.x * 256 + threadIdx.x;  // n*4096 + t
  const int n = id >> 12;
  const int t = id & (Tn - 1);
  const int f = idx[n & 255];
  const float c = (f == 0 ? 1.0f : 2.0f) * (1.0f / (float)Tn);
  const int p = (f * t) & (Tn - 1);
  const float th = (float)p * TWO_PI_OVER_T;
  ib[id] = (n < Mn) ? c * cosf(th) : -c * sinf(th);
}

// ---------------- forward DFT GEMM ------------------------------------------
// Y[r][n] = sum_t X[r][t] * fb[t][n], r=(b,h,d); a 64-row block is one (b,h) slab.
// grid (512/64=8, B*H=128), block 256 (8 waves, each a 16x32 C tile).
__global__ __launch_bounds__(256) void dft_fwd_kernel(const float* __restrict__ x,
                                                      const float* __restrict__ fb,
                                                      float* __restrict__ yre,
                                                      float* __restrict__ yim) {
  __shared__ float smem[2 * 64 * 68];
  float* As = smem;
  float* Bs = smem + 64 * 68;
  const int n0 = blockIdx.x * 64;
  const int bh = blockIdx.y;
  const int b = bh >> 4, h = bh & 15;
  const float* xb = x + (size_t)b * (Tn * Hn * Dn) + (size_t)h * Dn;
  const float* fbn = fb + n0;
  const int w = threadIdx.x >> 5, wrow = w >> 1, wc = w & 1;
  v8f c0 = ZERO8, c1 = ZERO8;

  F4x4 ra = fetch_64(xb, Hn * Dn);           // [t][d] tile kt=0
  F4x4 rb = fetch_64(fbn, 2 * Mn);

  for (int kt = 0; kt < Tn / 64; ++kt) {
    stash64_direct(As, ra);
    stash64_direct(Bs, rb);
    __syncthreads();
    if (kt + 1 < Tn / 64) {                   // overlap next-tile loads w/ WMMA
      ra = fetch_64(xb + (size_t)(kt + 1) * 64 * (Hn * Dn), Hn * Dn);
      rb = fetch_64(fbn + (size_t)(kt + 1) * 64 * (2 * Mn), 2 * Mn);
    }
    if (kt + 2 < Tn / 64) {                   // L2 prefetch of the HBM stream
      const int t = threadIdx.x;
      __builtin_prefetch(xb + (size_t)(kt + 2) * 64 * (Hn * Dn) +
                             (size_t)(t >> 2) * (Hn * Dn) + (t & 3) * 16, 0, 1);
    }
#pragma unroll
    for (int kk = 0; kk < 64; kk += 4) {
      const v2f a  = frag_ld(As, kk, 16 * wrow);
      const v2f b0 = frag_ld(Bs, kk, wc * 32);
      const v2f b1 = frag_ld(Bs, kk, wc * 32 + 16);
      c0 = wmma4(a, b0, c0);
      c1 = wmma4(a, b1, c1);
    }
    __syncthreads();
  }

  float* plane = (n0 < Mn) ? yre : yim;   // 64-col tile never straddles planes
  const int lane = threadIdx.x & 31, half = lane >> 4, lm = lane & 15;
  const int r0 = bh * 64;
  const int colb = (n0 & (Mn - 1)) + wc * 32;
#pragma unroll
  for (int nt = 0; nt < 2; ++nt)
#pragma unroll
    for (int rr = 0; rr < 8; ++rr) {
      const int row = r0 + 16 * wrow + rr + half * 8;
      plane[(size_t)row * Mn + colb + nt * 16 + lm] = nt ? c1[rr] : c0[rr];
    }
}

// ---------------- batched complex GEMM (attn and xqkv) ----------------------
// C[m][n] = sum_k A[m][k]*B[k][n] (complex, no conjugation).
// ADIR: A source is [k][m-contig]; else [m][k-contig]. Same for BDIR with n.
// F32 WMMA forbids A/B negation -> re uses a separate negative accumulator.
template <bool ADIR, bool BDIR, bool TANH>
__global__ __launch_bounds__(256) void cgemm_kernel(
    const float* __restrict__ Ar, const float* __restrict__ Ai, int ldA, long sA,
    const float* __restrict__ Br, const float* __restrict__ Bi, int ldB, long sB,
    float* __restrict__ Cr, float* __restrict__ Ci, int ldC, long sC, int K) {
  __shared__ float smem[4 * 32 * 68];
  float* Asr = smem;
  float* Asi = smem + 1 * 32 * 68;
  float* Bsr = smem + 2 * 32 * 68;
  float* Bsi = smem + 3 * 32 * 68;
  const int n0 = blockIdx.x * 64;
  const int m0 = blockIdx.y * 64;
  const long bt = blockIdx.z;
  const float* ar = Ar + bt * sA;
  const float* ai = Ai + bt * sA;
  const float* br = Br + bt * sB;
  const float* bi = Bi + bt * sB;
  const int w = threadIdx.x >> 5, wrow = w >> 1, wc = w & 1;
  v8f cp0 = ZERO8, cn0 = ZERO8, cI0 = ZERO8;
  v8f cp1 = ZERO8, cn1 = ZERO8, cI1 = ZERO8;

  auto fA = [&](const float* p, int k0) {
    return ADIR ? fetch_32_direct(p + (size_t)k0 * ldA + m0, ldA)
                : fetch_32_trans(p + (size_t)m0 * ldA + k0, ldA);
  };
  auto fB = [&](const float* p, int k0) {
    return BDIR ? fetch_32_direct(p + (size_t)k0 * ldB + n0, ldB)
                : fetch_32_trans(p + (size_t)n0 * ldB + k0, ldB);
  };

  F4x2 rar = fA(ar, 0), rai = fA(ai, 0), rbr = fB(br, 0), rbi = fB(bi, 0);

  for (int k0 = 0; k0 < K; k0 += 32) {
    if (ADIR) { stash32_direct(Asr, rar); stash32_direct(Asi, rai); }
    else      { stash32_trans(Asr, rar);  stash32_trans(Asi, rai);  }
    if (BDIR) { stash32_direct(Bsr, rbr); stash32_direct(Bsi, rbi); }
    else      { stash32_trans(Bsr, rbr);  stash32_trans(Bsi, rbi);  }
    __syncthreads();
    if (k0 + 32 < K) {                      // overlap next-tile loads w/ WMMA
      rar = fA(ar, k0 + 32); rai = fA(ai, k0 + 32);
      rbr = fB(br, k0 + 32); rbi = fB(bi, k0 + 32);
    }
#pragma unroll
    for (int kk = 0; kk < 32; kk += 4) {
      const v2f far = frag_ld(Asr, kk, 16 * wrow);
      const v2f fai = frag_ld(Asi, kk, 16 * wrow);
      const v2f br0 = frag_ld(Bsr, kk, wc * 32);
      const v2f bi0 = frag_ld(Bsi, kk, wc * 32);
      const v2f br1 = frag_ld(Bsr, kk, wc * 32 + 16);
      const v2f bi1 = frag_ld(Bsi, kk, wc * 32 + 16);
      cp0 = wmma4(far, br0, cp0);   // + ar*br
      cn0 = wmma4(fai, bi0, cn0);   // + ai*bi  (subtracted later)
      cI0 = wmma4(far, bi0, cI0);   // + ar*bi
      cI0 = wmma4(fai, br0, cI0);   // + ai*br
      cp1 = wmma4(far, br1, cp1);
      cn1 = wmma4(fai, bi1, cn1);
      cI1 = wmma4(far, bi1, cI1);
      cI1 = wmma4(fai, br1, cI1);
    }
    __syncthreads();
  }

  const int lane = threadIdx.x & 31, half = lane >> 4, lm = lane & 15;
  float* crb = Cr + bt * sC;
  float* cib = Ci + bt * sC;
#pragma unroll
  for (int nt = 0; nt < 2; ++nt)
#pragma unroll
    for (int rr = 0; rr < 8; ++rr) {
      float re = nt ? (cp1[rr] - cn1[rr]) : (cp0[rr] - cn0[rr]);
      float im = nt ? cI1[rr] : cI0[rr];
      if (TANH) {
        // complex tanh: (sinh 2a + i sin 2b) / (cosh 2a + cos 2b), saturate big |a|
        if (fabsf(re) > 20.0f) {
          re = re > 0.0f ? 1.0f : -1.0f;
          im = 0.0f;
        } else {
          const float den = coshf(2.0f * re) + cosf(2.0f * im);
          const float nr = sinhf(2.0f * re);
          const float ni = sinf(2.0f * im);
          re = nr / den;
          im = ni / den;
        }
      }
      const int row = m0 + 16 * wrow + rr + half * 8;
      const int col = n0 + wc * 32 + nt * 16 + lm;
      crb[(size_t)row * ldC + col] = re;
      cib[(size_t)row * ldC + col] = im;
    }
}

// ---------------- mode mixing with w (per-(h,x) 8x64x64 complex) ------------
// out[b,h,o,x] = SCALE^2 * sum_d xqkv[b,h,d,x] * (wr+i*wi)[h,d,o,x]
// Thread per (h,o,x); x contiguous across lanes -> fully coalesced w/xqkv loads.
__global__ __launch_bounds__(256) void wmix_kernel(const float* __restrict__ xr,
                                                   const float* __restrict__ xi,
                                                   const float* __restrict__ wr,
                                                   const float* __restrict__ wi,
                                                   float* __restrict__ or_,
                                                   float* __restrict__ oi_) {
  const int id = blockIdx.x * 256 + threadIdx.x;  // h*64*256 + o*256 + x
  const int x = id & (Mn - 1);
  const int o = (id >> 8) & (Dn - 1);
  const int h = id >> 14;
  float aR[Bn], aI[Bn];
#pragma unroll
  for (int b = 0; b < Bn; ++b) { aR[b] = 0.f; aI[b] = 0.f; }

  size_t wp = (((size_t)h * Dn) * Dn + o) * Mn + x;        // (h,d=0,o,x)
  const size_t xbase = ((size_t)h * Dn) * Mn + x;          // (b=0,h,d=0,x)
  for (int d = 0; d < Dn; ++d) {
    const float cwr = wr[wp];
    const float cwi = wi[wp];
    wp += (size_t)Dn * Mn;
    const size_t xp0 = xbase + (size_t)d * Mn;
#pragma unroll
    for (int b = 0; b < Bn; ++b) {
      const size_t xp = xp0 + (size_t)b * (Hn * Dn * Mn);
      const float vr = xr[xp];
      const float vi = xi[xp];
      aR[b] += vr * cwr - vi * cwi;
      aI[b] += vr * cwi + vi * cwr;
    }
  }
#pragma unroll
  for (int b = 0; b < Bn; ++b) {
    const size_t op = (((size_t)b * Hn + h) * Dn + o) * Mn + x;
    or_[op] = aR[b] * SCALE2;
    oi_[op] = aI[b] * SCALE2;
  }
}

// ---------------- inverse DFT GEMM ------------------------------------------
// out[r=(b,h,o)][t] = sum_{n<512} F[r][n] * ib[n][t]; C tile transposed through
// LDS so (B,T,H,D)-layout stores are float4-coalesced. grid (T/64=64, B*H=128).
__global__ __launch_bounds__(256) void idft_kernel(const float* __restrict__ fre,
                                                   const float* __restrict__ fim,
                                                   const float* __restrict__ ib,
                                                   float* __restrict__ out) {
  __shared__ float As[64 * 68];
  __shared__ float Bs[64 * 68];
  __shared__ float Ct[64 * 65];  // [t][o], padded
  const int n0 = blockIdx.x * 64;  // t tile
  const int bh = blockIdx.y;
  const int b = bh >> 4, h = bh & 15;
  const int r0 = bh * 64;
  const int w = threadIdx.x >> 5, wrow = w >> 1, wc = w & 1;
  v8f c0 = ZERO8, c1 = ZERO8;

  auto fA = [&](int k0) {
    const float* plane = (k0 < Mn) ? fre : fim;  // k-tile never straddles planes
    return fetch_64(plane + (size_t)r0 * Mn + (k0 & (Mn - 1)), Mn);  // [m=o][k]
  };

  F4x4 ra = fA(0);
  F4x4 rb = fetch_64(ib + n0, Tn);

  for (int kt = 0; kt < 8; ++kt) {
    stash64_trans(As, ra);
    stash64_direct(Bs, rb);
    __syncthreads();
    if (kt + 1 < 8) {
      ra = fA((kt + 1) * 64);
      rb = fetch_64(ib + (size_t)(kt + 1) * 64 * Tn + n0, Tn);
    }
#pragma unroll
    for (int kk = 0; kk < 64; kk += 4) {
      const v2f a  = frag_ld(As, kk, 16 * wrow);
      const v2f b0 = frag_ld(Bs, kk, wc * 32);
      const v2f b1 = frag_ld(Bs, kk, wc * 32 + 16);
      c0 = wmma4(a, b0, c0);
      c1 = wmma4(a, b1, c1);
    }
    __syncthreads();
  }

  const int lane = threadIdx.x & 31, half = lane >> 4, lm = lane & 15;
#pragma unroll
  for (int nt = 0; nt < 2; ++nt)
#pragma unroll
    for (int rr = 0; rr < 8; ++rr) {
      const int ol = 16 * wrow + rr + half * 8;
      const int tl = wc * 32 + nt * 16 + lm;
      Ct[tl * 65 + ol] = nt ? c1[rr] : c0[rr];
    }
  __syncthreads();

  float* ob = out + ((size_t)b * Tn + n0) * (Hn * Dn) + (size_t)h * Dn;
  const int t = threadIdx.x;
#pragma unroll
  for (int i = 0; i < 4; ++i) {
    const int tl = i * 16 + (t >> 4);
    const int oq = (t & 15) << 2;
    float4 v;
    v.x = Ct[tl * 65 + oq + 0];
    v.y = Ct[tl * 65 + oq + 1];
    v.z = Ct[tl * 65 + oq + 2];
    v.w = Ct[tl * 65 + oq + 3];
    *(float4*)(ob + (size_t)tl * (Hn * Dn) + oq) = v;
  }
}

// ---------------- host launcher ---------------------------------------------

extern "C" void kernel_launch(void* const* d_in, const int* in_sizes, int n_in,
                              void* d_out, int out_size, void* d_ws, size_t ws_size,
                              hipStream_t stream) {
  (void)in_sizes; (void)n_in; (void)out_size; (void)ws_size;
  const float* q   = (const float*)d_in[0];
  const float* k   = (const float*)d_in[1];
  const float* v   = (const float*)d_in[2];
  const float* wre = (const float*)d_in[3];
  const float* wim = (const float*)d_in[4];
  const int* idxq  = (const int*)d_in[5];
  const int* idxkv = (const int*)d_in[6];
  float* out = (float*)d_out;

  float* ws = (float*)d_ws;
  size_t cur = 0;
  auto alloc = [&](size_t n) { float* p = ws + cur; cur += n; return p; };
  const size_t FB    = (size_t)Tn * 2 * Mn;        // 2.10 M floats
  const size_t PLANE = (size_t)Bn * Hn * Dn * Mn;  // 2.10 M floats
  const size_t ATT   = (size_t)Bn * Hn * Mn * Mn;  // 8.39 M floats

  float* fbq  = alloc(FB);
  float* fbkv = alloc(FB);
  float* ibs  = alloc(FB);
  float* xqre = alloc(PLANE); float* xqim = alloc(PLANE);
  float* xkre = alloc(PLANE); float* xkim = alloc(PLANE);
  float* xvre = alloc(PLANE); float* xvim = alloc(PLANE);
  float* atre = alloc(ATT);   float* atim = alloc(ATT);
  float* qvre = alloc(PLANE); float* qvim = alloc(PLANE);
  float* xwre = alloc(PLANE); float* xwim = alloc(PLANE);

  // 1. twiddle bases (exact integer phase reduction mod T)
  build_fbasis_kernel<<<FB / 256, 256, 0, stream>>>(idxq, fbq);
  build_fbasis_kernel<<<FB / 256, 256, 0, stream>>>(idxkv, fbkv);
  build_ibasis_kernel<<<FB / 256, 256, 0, stream>>>(idxq, ibs);

  // 2. mode-restricted forward DFT of q,k,v  (WMMA f32 GEMM)
  dft_fwd_kernel<<<dim3(8, Bn * Hn), 256, 0, stream>>>(q, fbq, xqre, xqim);
  dft_fwd_kernel<<<dim3(8, Bn * Hn), 256, 0, stream>>>(k, fbkv, xkre, xkim);
  dft_fwd_kernel<<<dim3(8, Bn * Hn), 256, 0, stream>>>(v, fbkv, xvre, xvim);

  // 3. attn = tanh(Xq^T Xk): A from [d][x] (direct), B from [d][y] (direct)
  cgemm_kernel<true, true, true><<<dim3(4, 4, Bn * Hn), 256, 0, stream>>>(
      xqre, xqim, Mn, (long)(Dn * Mn),
      xkre, xkim, Mn, (long)(Dn * Mn),
      atre, atim, Mn, (long)(Mn * Mn), Dn);

  // 4. xqkv = Xv * attn^T: A from [d][y] (m-major -> trans), B from [x][y] (trans)
  cgemm_kernel<false, false, false><<<dim3(4, 1, Bn * Hn), 256, 0, stream>>>(
      xvre, xvim, Mn, (long)(Dn * Mn),
      atre, atim, Mn, (long)(Mn * Mn),
      qvre, qvim, Mn, (long)(Dn * Mn), Mn);

  // 5. mode mixing with w (+ fold SCALE^2)
  wmix_kernel<<<(Hn * Dn * Mn) / 256, 256, 0, stream>>>(qvre, qvim, wre, wim,
                                                        xwre, xwim);

  // 6. scatter + irfft == inverse DFT GEMM over selected modes
  idft_kernel<<<dim3(Tn / 64, Bn * Hn), 256, 0, stream>>>(xwre, xwim, ibs, out);
}